// BioV_19748259627109
// MI455X (gfx1250) — compile-verified
//
#include <hip/hip_runtime.h>
#include <math.h>

// ---------------- problem constants ----------------
#define BB   8
#define CC   3
#define TT   32
#define HH   128
#define WW   128
#define HWSZ (HH*WW)            // 16384
#define NPC  (TT*HWSZ)          // 524288 elements per (b,c)
#define NBC  (BB*CC)            // 24
#define YSZ  (NBC*NPC)          // 12,582,912 floats

// workspace layout (float offsets)
#define OFF_Y     0
#define OFF_SUM   (YSZ)               // 24
#define OFF_SQ    (YSZ+24)            // 24
#define OFF_KVT   (YSZ+48)            // 24*2*32 = 1536 (zeroed, atomics)
#define OFF_NA    (YSZ+1584)          // 24
#define OFF_NB    (YSZ+1608)          // 24
#define OFF_SMAX  (YSZ+1632)          // 24
#define OFF_SSUM  (YSZ+1656)          // 24
#define OFF_PT    (YSZ+1680)          // 24*32 = 768
#define OFF_KVS   (YSZ+2448)          // 24*2*16384 = 786432

typedef __attribute__((ext_vector_type(2))) float v2f;
typedef __attribute__((ext_vector_type(8))) float v8f;

#if defined(__HIP_DEVICE_COMPILE__) && __has_builtin(__builtin_amdgcn_wmma_f32_16x16x4_f32)
#define USE_WMMA 1
#else
#define USE_WMMA 0
#endif

__device__ __forceinline__ float silu_f(float v) {
    return v / (1.0f + __expf(-v));
}

// ---------------- K0: zero accumulators ----------------
__global__ void k_zero(float* __restrict__ p, int n) {
    int i = blockIdx.x * 256 + threadIdx.x;
    if (i < n) p[i] = 0.0f;
}

// ---------------- K1: spatial 3x3 conv (1->3 ch) + bias + SiLU ----------------
__global__ __launch_bounds__(256)
void k_spatial(const float* __restrict__ x, const float* __restrict__ wsp,
               const float* __restrict__ bsp, float* __restrict__ y) {
    __shared__ float sw[27];
    __shared__ float sb[3];
    int tid = threadIdx.x;
    if (tid < 27) sw[tid] = wsp[tid];
    if (tid < 3)  sb[tid] = bsp[tid];
    __syncthreads();

    int idx = blockIdx.x * 256 + tid;             // over [B,T,HW]
    int b  = idx / (TT * HWSZ);
    int r  = idx - b * (TT * HWSZ);
    int t  = r / HWSZ;
    int hw = r - t * HWSZ;
    int h  = hw >> 7;
    int w  = hw & 127;

    const float* xb = x + (size_t)(b * TT + t) * HWSZ;
    float acc0 = 0.f, acc1 = 0.f, acc2 = 0.f;
    #pragma unroll
    for (int di = -1; di <= 1; ++di) {
        #pragma unroll
        for (int dj = -1; dj <= 1; ++dj) {
            int hh = h + di, ww2 = w + dj;
            float xv = (hh >= 0 && hh < HH && ww2 >= 0 && ww2 < WW)
                       ? xb[hh * WW + ww2] : 0.0f;
            int k = (di + 1) * 3 + (dj + 1);
            acc0 += sw[0 * 9 + k] * xv;
            acc1 += sw[1 * 9 + k] * xv;
            acc2 += sw[2 * 9 + k] * xv;
        }
    }
    size_t base = ((size_t)(b * CC) * TT + t) * HWSZ + hw;
    y[base]                      = silu_f(acc0 + sb[0]);
    y[base + (size_t)TT * HWSZ]  = silu_f(acc1 + sb[1]);
    y[base + (size_t)2*TT*HWSZ]  = silu_f(acc2 + sb[2]);
}

// ---------------- K2: temporal 7-tap depthwise conv + SiLU + SiLU + stats ----------------
__global__ __launch_bounds__(256)
void k_temporal(const float* __restrict__ y, const float* __restrict__ wtp,
                const float* __restrict__ btp, float* __restrict__ g,
                float* __restrict__ sumAcc, float* __restrict__ sqAcc) {
    __shared__ float rs[256];
    __shared__ float rq[256];
    int tid = threadIdx.x;
    int bc  = blockIdx.y;
    int c   = bc % CC;
    int i   = blockIdx.x * 256 + tid;             // over [T,HW]
    int t   = i / HWSZ;
    int hw  = i - t * HWSZ;

    float wk[7];
    #pragma unroll
    for (int k = 0; k < 7; ++k) wk[k] = wtp[c * 7 + k];

    const float* yb = y + (size_t)bc * NPC;
    float z = 0.f;
    #pragma unroll
    for (int k = 0; k < 7; ++k) {
        int tt = t + k - 3;
        if (tt >= 0 && tt < TT) z += yb[(size_t)tt * HWSZ + hw] * wk[k];
    }
    float gv = silu_f(silu_f(z + btp[c]));
    g[(size_t)bc * NPC + i] = gv;

    rs[tid] = gv;
    rq[tid] = gv * gv;
    __syncthreads();
    for (int s = 128; s > 0; s >>= 1) {
        if (tid < s) { rs[tid] += rs[tid + s]; rq[tid] += rq[tid + s]; }
        __syncthreads();
    }
    if (tid == 0) {
        atomicAdd(&sumAcc[bc], rs[0]);
        atomicAdd(&sqAcc[bc],  rq[0]);
    }
}

// ---------------- K3: SwitchNorm stat finalize -> per-(b,c) affine ----------------
__global__ void k_stats(const float* __restrict__ sumAcc, const float* __restrict__ sqAcc,
                        const float* __restrict__ mwIn, const float* __restrict__ vwIn,
                        const float* __restrict__ snw, const float* __restrict__ snb,
                        float* __restrict__ normA, float* __restrict__ normB) {
    __shared__ float s_mean[NBC];
    __shared__ float s_var[NBC];
    __shared__ float s_temp[NBC];
    int tid = threadIdx.x;
    const float N = (float)NPC;
    if (tid < NBC) {
        float s = sumAcc[tid], q = sqAcc[tid];
        float m = s / N;
        float v = (q - s * m) / (N - 1.0f);       // ddof=1
        s_mean[tid] = m;
        s_var[tid]  = v;
        s_temp[tid] = v + m * m;
    }
    __syncthreads();
    if (tid == 0) {
        // softmax(mean_weight), softmax(var_weight)
        float mw[3], vw[3];
        float mm = fmaxf(fmaxf(mwIn[0], mwIn[1]), mwIn[2]);
        float vm = fmaxf(fmaxf(vwIn[0], vwIn[1]), vwIn[2]);
        float ms = 0.f, vs = 0.f;
        for (int k = 0; k < 3; ++k) { mw[k] = expf(mwIn[k] - mm); ms += mw[k]; }
        for (int k = 0; k < 3; ++k) { vw[k] = expf(vwIn[k] - vm); vs += vw[k]; }
        for (int k = 0; k < 3; ++k) { mw[k] /= ms; vw[k] /= vs; }

        float mean_ln[BB], var_ln[BB], mean_bn[CC], var_bn[CC];
        for (int b = 0; b < BB; ++b) {
            float am = 0.f, at = 0.f;
            for (int c = 0; c < CC; ++c) { am += s_mean[b*CC+c]; at += s_temp[b*CC+c]; }
            am /= CC; at /= CC;
            mean_ln[b] = am; var_ln[b] = at - am * am;
        }
        for (int c = 0; c < CC; ++c) {
            float am = 0.f, at = 0.f;
            for (int b = 0; b < BB; ++b) { am += s_mean[b*CC+c]; at += s_temp[b*CC+c]; }
            am /= BB; at /= BB;
            mean_bn[c] = am; var_bn[c] = at - am * am;
        }
        for (int bc = 0; bc < NBC; ++bc) {
            int b = bc / CC, c = bc % CC;
            float mean = mw[0]*s_mean[bc] + mw[1]*mean_ln[b] + mw[2]*mean_bn[c];
            float var  = vw[0]*s_var[bc]  + vw[1]*var_ln[b]  + vw[2]*var_bn[c];
            float rstd = rsqrtf(var + 1e-5f);
            float A = rstd * snw[c];
            normA[bc] = A;
            normB[bc] = snb[c] - mean * A;
        }
    }
}

// ---------------- K4: kv_s / kv_t contractions (FP32 WMMA) ----------------
// one wave per (b,c, 64-hw chunk); xn tile (32 t x 64 hw) staged in LDS.
__global__ __launch_bounds__(32)
void k_kv(const float* __restrict__ g, const float* __restrict__ wkvs,
          const float* __restrict__ wkvt, const float* __restrict__ normA,
          const float* __restrict__ normB, float* __restrict__ kvs,
          float* __restrict__ kvt) {
    __shared__ float xt[TT][68];                  // pad 64->68 (bank spread)
    const int lane = threadIdx.x;
    const int bc   = blockIdx.y;
    const int hw0  = blockIdx.x * 64;
    const float A  = normA[bc];
    const float Bv = normB[bc];
    const float* gb = g + (size_t)bc * NPC + hw0;
    for (int t = 0; t < TT; ++t) {
        xt[t][lane]      = gb[(size_t)t * HWSZ + lane]      * A + Bv;
        xt[t][lane + 32] = gb[(size_t)t * HWSZ + lane + 32] * A + Bv;
    }
    __syncthreads();

    const int mm = lane & 15;
    const int hi = lane >> 4;                     // 0: K{0,1}, 1: K{2,3}

#if USE_WMMA
    // ---- kv_s[o,hw] = sum_t wkvs[o,t]*xn[t,hw] : [16pad x 32]*[32 x 16] ----
    v2f afs[8];
    #pragma unroll
    for (int s = 0; s < 8; ++s) {
        int k0 = s * 4 + hi * 2;
        afs[s].x = (mm < 2) ? wkvs[mm * TT + k0]     : 0.0f;
        afs[s].y = (mm < 2) ? wkvs[mm * TT + k0 + 1] : 0.0f;
    }
    #pragma unroll
    for (int nt = 0; nt < 4; ++nt) {
        v8f acc = {};
        #pragma unroll
        for (int s = 0; s < 8; ++s) {
            int k0 = s * 4 + hi * 2;
            v2f bf;
            bf.x = xt[k0][nt * 16 + mm];
            bf.y = xt[k0 + 1][nt * 16 + mm];
            acc = __builtin_amdgcn_wmma_f32_16x16x4_f32(
                false, afs[s], false, bf, (short)0, acc, false, false);
        }
        if (lane < 16) {
            kvs[(size_t)(bc * 2 + 0) * HWSZ + hw0 + nt * 16 + lane] = acc[0]; // row 0
            kvs[(size_t)(bc * 2 + 1) * HWSZ + hw0 + nt * 16 + lane] = acc[1]; // row 1
        }
    }
    // ---- kv_t[o,t] += sum_hw wkvt[o,hw]*xn[t,hw] : [16pad x 64]*[64 x 16]x2 ----
    v8f a0 = {}; v8f a1 = {};
    #pragma unroll 4
    for (int kk = 0; kk < 16; ++kk) {
        int k0 = kk * 4 + hi * 2;
        v2f at;
        at.x = (mm < 2) ? wkvt[(size_t)mm * HWSZ + hw0 + k0]     : 0.0f;
        at.y = (mm < 2) ? wkvt[(size_t)mm * HWSZ + hw0 + k0 + 1] : 0.0f;
        v2f b0, b1;
        b0.x = xt[mm][k0];      b0.y = xt[mm][k0 + 1];
        b1.x = xt[mm + 16][k0]; b1.y = xt[mm + 16][k0 + 1];
        a0 = __builtin_amdgcn_wmma_f32_16x16x4_f32(false, at, false, b0, (short)0, a0, false, false);
        a1 = __builtin_amdgcn_wmma_f32_16x16x4_f32(false, at, false, b1, (short)0, a1, false, false);
    }
    if (lane < 16) {
        atomicAdd(&kvt[(bc * 2 + 0) * TT + lane],      a0[0]);
        atomicAdd(&kvt[(bc * 2 + 1) * TT + lane],      a0[1]);
        atomicAdd(&kvt[(bc * 2 + 0) * TT + 16 + lane], a1[0]);
        atomicAdd(&kvt[(bc * 2 + 1) * TT + 16 + lane], a1[1]);
    }
#else
    // scalar fallback (host pass / missing builtin)
    for (int rep = 0; rep < 2; ++rep) {
        int col = lane + rep * 32;
        float s0 = 0.f, s1 = 0.f;
        for (int t = 0; t < TT; ++t) {
            float v = xt[t][col];
            s0 += wkvs[t] * v;
            s1 += wkvs[TT + t] * v;
        }
        kvs[(size_t)(bc * 2 + 0) * HWSZ + hw0 + col] = s0;
        kvs[(size_t)(bc * 2 + 1) * HWSZ + hw0 + col] = s1;
    }
    {
        float s0 = 0.f, s1 = 0.f;
        for (int col = 0; col < 64; ++col) {
            float v = xt[lane][col];
            s0 += wkvt[hw0 + col] * v;
            s1 += wkvt[(size_t)HWSZ + hw0 + col] * v;
        }
        atomicAdd(&kvt[(bc * 2 + 0) * TT + lane], s0);
        atomicAdd(&kvt[(bc * 2 + 1) * TT + lane], s1);
    }
#endif
    (void)mm; (void)hi;
}

// ---------------- K5: softmax reductions ----------------
__global__ __launch_bounds__(256)
void k_softmax(const float* __restrict__ kvs, const float* __restrict__ kvt,
               float* __restrict__ smax, float* __restrict__ ssum,
               float* __restrict__ pt) {
    __shared__ float red[256];
    __shared__ float s_mx;
    int tid = threadIdx.x;
    int bc  = blockIdx.x;
    const float* key = kvs + (size_t)(bc * 2) * HWSZ;

    float m = -3.4e38f;
    for (int i = tid; i < HWSZ; i += 256) m = fmaxf(m, key[i]);
    red[tid] = m;
    __syncthreads();
    for (int s = 128; s > 0; s >>= 1) {
        if (tid < s) red[tid] = fmaxf(red[tid], red[tid + s]);
        __syncthreads();
    }
    if (tid == 0) s_mx = red[0];
    __syncthreads();
    float mx = s_mx;

    float acc = 0.f;
    for (int i = tid; i < HWSZ; i += 256) acc += expf(key[i] - mx);
    red[tid] = acc;
    __syncthreads();
    for (int s = 128; s > 0; s >>= 1) {
        if (tid < s) red[tid] += red[tid + s];
        __syncthreads();
    }
    if (tid == 0) {
        smax[bc] = mx;
        ssum[bc] = red[0];
        // temporal softmax over 32 t (serial, trivial)
        const float* kt_ = kvt + (bc * 2 + 0) * TT;
        const float* vt_ = kvt + (bc * 2 + 1) * TT;
        float m2 = -3.4e38f;
        for (int t = 0; t < TT; ++t) m2 = fmaxf(m2, kt_[t]);
        float s2 = 0.f;
        for (int t = 0; t < TT; ++t) s2 += expf(kt_[t] - m2);
        for (int t = 0; t < TT; ++t) {
            float kt = expf(kt_[t] - m2) / s2;
            pt[bc * TT + t] = sqrtf(kt) * vt_[t];
        }
    }
}

// ---------------- K6: out = (sqrt(kt)*vt)[t] * (sqrt(ks)*vs)[hw] ----------------
__global__ __launch_bounds__(256)
void k_final(const float* __restrict__ kvs, const float* __restrict__ smax,
             const float* __restrict__ ssum, const float* __restrict__ pt,
             float* __restrict__ out) {
    __shared__ float s_pt[TT];
    int tid = threadIdx.x;
    int bc  = blockIdx.y;
    if (tid < TT) s_pt[tid] = pt[bc * TT + tid];
    __syncthreads();
    int hw = blockIdx.x * 256 + tid;
    float key = kvs[(size_t)(bc * 2 + 0) * HWSZ + hw];
    float val = kvs[(size_t)(bc * 2 + 1) * HWSZ + hw];
    float ks  = expf(key - smax[bc]) / ssum[bc];
    float ps  = sqrtf(ks) * val;
    float* ob = out + (size_t)bc * NPC + hw;
    #pragma unroll 4
    for (int t = 0; t < TT; ++t) ob[(size_t)t * HWSZ] = s_pt[t] * ps;
}

// ---------------- launch ----------------
extern "C" void kernel_launch(void* const* d_in, const int* in_sizes, int n_in,
                              void* d_out, int out_size, void* d_ws, size_t ws_size,
                              hipStream_t stream) {
    const float* x    = (const float*)d_in[0];
    const float* wsp  = (const float*)d_in[1];
    const float* bsp  = (const float*)d_in[2];
    const float* wtp  = (const float*)d_in[3];
    const float* btp  = (const float*)d_in[4];
    const float* snw  = (const float*)d_in[5];
    const float* snb  = (const float*)d_in[6];
    const float* mwI  = (const float*)d_in[7];
    const float* vwI  = (const float*)d_in[8];
    const float* wkvs = (const float*)d_in[9];
    const float* wkvt = (const float*)d_in[10];
    float* out = (float*)d_out;
    float* ws  = (float*)d_ws;

    float* y     = ws + OFF_Y;
    float* sumA  = ws + OFF_SUM;
    float* sqA   = ws + OFF_SQ;
    float* kvt   = ws + OFF_KVT;
    float* nA    = ws + OFF_NA;
    float* nB    = ws + OFF_NB;
    float* smax  = ws + OFF_SMAX;
    float* ssumA = ws + OFF_SSUM;
    float* pt    = ws + OFF_PT;
    float* kvs   = ws + OFF_KVS;
    float* g     = out;   // reuse d_out as g scratch; fully overwritten by k_final

    k_zero<<<(1584 + 255) / 256, 256, 0, stream>>>(sumA, 1584);
    k_spatial<<<(BB * TT * HWSZ) / 256, 256, 0, stream>>>(x, wsp, bsp, y);
    k_temporal<<<dim3(NPC / 256, NBC), 256, 0, stream>>>(y, wtp, btp, g, sumA, sqA);
    k_stats<<<1, 32, 0, stream>>>(sumA, sqA, mwI, vwI, snw, snb, nA, nB);
    k_kv<<<dim3(HWSZ / 64, NBC), 32, 0, stream>>>(g, wkvs, wkvt, nA, nB, kvs, kvt);
    k_softmax<<<NBC, 256, 0, stream>>>(kvs, kvt, smax, ssumA, pt);
    k_final<<<dim3(HWSZ / 256, NBC), 256, 0, stream>>>(kvs, smax, ssumA, pt, out);
}